// DiscretizedSSM_13005160972580
// MI455X (gfx1250) — compile-verified
//
#include <hip/hip_runtime.h>

typedef float v2f __attribute__((ext_vector_type(2)));
typedef float v8f __attribute__((ext_vector_type(8)));

// Problem constants (from reference setup)
#define BSZ   16
#define SEQ   4096
#define DM    256
#define CHL   32          // chunk length
#define NCH   128         // SEQ / CHL
#define MATN  65536       // 256*256 elements

static __device__ __forceinline__ v8f wmma4(v2f a, v2f b, v8f c) {
    // D = A(16x4 f32) x B(4x16 f32) + C(16x16 f32)
    return __builtin_amdgcn_wmma_f32_16x16x4_f32(
        false, a, false, b, (short)0, c, false, false);
}

// B-fragment load from pair-interleaved ("swizzled") weight matrix:
// Wsw[((k>>1)*256 + n)*2 + (k&1)] == W[k][n]  -> one aligned 8B load per frag.
static __device__ __forceinline__ v2f ldb_sw(const float* __restrict__ W,
                                             int kp, int col) {
    return ((const v2f*)W)[(size_t)kp * 256 + col];
}

// ---------------------------------------------------------------------------
// Row-major-Q GEMM body (stage 0 only): Out[rows,256] = alpha*(P@Q) + beta*R
// ---------------------------------------------------------------------------
static __device__ __forceinline__ void gemm_rows32(
    float* __restrict__ Out, const float* __restrict__ P,
    const float* __restrict__ Q, const float* __restrict__ R,
    float alpha, float beta, int rowBlock)
{
    const int tid  = threadIdx.x;
    const int wave = tid >> 5;
    const int lane = tid & 31;
    const int half = lane >> 4;
    const int lr   = lane & 15;
    const int r0   = rowBlock * 32 + (wave >> 2) * 16;
    const int n0   = (wave & 3) * 64;

    v8f acc0 = {}, acc1 = {}, acc2 = {}, acc3 = {};
    const float* prow = P + (size_t)(r0 + lr) * 256;

#pragma unroll 4
    for (int k0 = 0; k0 < 256; k0 += 4) {
        v2f a = *(const v2f*)(prow + k0 + 2 * half);
        const float* q = Q + (size_t)(k0 + 2 * half) * 256 + lr;
        v2f b0; b0.x = q[n0 +  0]; b0.y = q[256 + n0 +  0];
        v2f b1; b1.x = q[n0 + 16]; b1.y = q[256 + n0 + 16];
        v2f b2; b2.x = q[n0 + 32]; b2.y = q[256 + n0 + 32];
        v2f b3; b3.x = q[n0 + 48]; b3.y = q[256 + n0 + 48];
        acc0 = wmma4(a, b0, acc0);
        acc1 = wmma4(a, b1, acc1);
        acc2 = wmma4(a, b2, acc2);
        acc3 = wmma4(a, b3, acc3);
    }

#pragma unroll
    for (int g = 0; g < 8; ++g) {
        const int row = r0 + g + 8 * half;
        float* o = Out + (size_t)row * 256 + n0 + lr;
        if (R) {
            const float* r = R + (size_t)row * 256 + n0 + lr;
            o[ 0] = alpha * acc0[g] + beta * r[ 0];
            o[16] = alpha * acc1[g] + beta * r[16];
            o[32] = alpha * acc2[g] + beta * r[32];
            o[48] = alpha * acc3[g] + beta * r[48];
        } else {
            o[ 0] = alpha * acc0[g];
            o[16] = alpha * acc1[g];
            o[32] = alpha * acc2[g];
            o[48] = alpha * acc3[g];
        }
    }
}

__global__ void k_gemm256(float* Out, const float* P, const float* Q,
                          const float* R, float alpha, float beta) {
    gemm_rows32(Out, P, Q, R, alpha, beta, blockIdx.x);
}

// Pw[k+i] = Pw[i] @ Pw[k]   (powers of At by log-doubling), i = blockIdx.y+1
__global__ void k_powers(float* PwBase, int k) {
    const int i = blockIdx.y + 1;
    gemm_rows32(PwBase + (size_t)(k + i - 1) * MATN,
                PwBase + (size_t)(i - 1) * MATN,
                PwBase + (size_t)(k - 1) * MATN,
                nullptr, 1.0f, 0.0f, blockIdx.x);
}

// ---------------------------------------------------------------------------
// Swizzled-Q GEMM (stage 1): Out[rows,256] = P @ Qsw
// ---------------------------------------------------------------------------
__global__ void k_gemm256_sw(float* __restrict__ Out,
                             const float* __restrict__ P,
                             const float* __restrict__ Q) {
    const int tid  = threadIdx.x;
    const int wave = tid >> 5;
    const int lane = tid & 31;
    const int half = lane >> 4;
    const int lr   = lane & 15;
    const int r0   = blockIdx.x * 32 + (wave >> 2) * 16;
    const int n0   = (wave & 3) * 64;

    v8f acc0 = {}, acc1 = {}, acc2 = {}, acc3 = {};
    const float* prow = P + (size_t)(r0 + lr) * 256;

#pragma unroll 4
    for (int k0 = 0; k0 < 256; k0 += 4) {
        v2f a = *(const v2f*)(prow + k0 + 2 * half);
        const int kp = (k0 >> 1) + half;
        v2f b0 = ldb_sw(Q, kp, n0 + lr);
        v2f b1 = ldb_sw(Q, kp, n0 + lr + 16);
        v2f b2 = ldb_sw(Q, kp, n0 + lr + 32);
        v2f b3 = ldb_sw(Q, kp, n0 + lr + 48);
        acc0 = wmma4(a, b0, acc0);
        acc1 = wmma4(a, b1, acc1);
        acc2 = wmma4(a, b2, acc2);
        acc3 = wmma4(a, b3, acc3);
    }
#pragma unroll
    for (int g = 0; g < 8; ++g) {
        const int row = r0 + g + 8 * half;
        float* o = Out + (size_t)row * 256 + n0 + lr;
        o[ 0] = acc0[g]; o[16] = acc1[g]; o[32] = acc2[g]; o[48] = acc3[g];
    }
}

// y = x@Ct + u@Dt, in-place over xio (rows = b*SEQ + t); Ct/Dt swizzled
__global__ void k_output(float* xio, const float* __restrict__ u,
                         const float* __restrict__ Ct,
                         const float* __restrict__ Dt) {
    const int tid  = threadIdx.x;
    const int wave = tid >> 5;
    const int lane = tid & 31;
    const int half = lane >> 4;
    const int lr   = lane & 15;
    const int r0   = blockIdx.x * 32 + (wave >> 2) * 16;
    const int n0   = (wave & 3) * 64;

    v8f acc0 = {}, acc1 = {}, acc2 = {}, acc3 = {};
    const float* xrow = xio + (size_t)(r0 + lr) * 256;
    const float* urow = u   + (size_t)(r0 + lr) * 256;

#pragma unroll 4
    for (int k0 = 0; k0 < 256; k0 += 4) {
        v2f a = *(const v2f*)(xrow + k0 + 2 * half);
        const int kp = (k0 >> 1) + half;
        v2f b0 = ldb_sw(Ct, kp, n0 + lr);
        v2f b1 = ldb_sw(Ct, kp, n0 + lr + 16);
        v2f b2 = ldb_sw(Ct, kp, n0 + lr + 32);
        v2f b3 = ldb_sw(Ct, kp, n0 + lr + 48);
        acc0 = wmma4(a, b0, acc0);
        acc1 = wmma4(a, b1, acc1);
        acc2 = wmma4(a, b2, acc2);
        acc3 = wmma4(a, b3, acc3);
    }
#pragma unroll 4
    for (int k0 = 0; k0 < 256; k0 += 4) {
        v2f a = *(const v2f*)(urow + k0 + 2 * half);
        const int kp = (k0 >> 1) + half;
        v2f b0 = ldb_sw(Dt, kp, n0 + lr);
        v2f b1 = ldb_sw(Dt, kp, n0 + lr + 16);
        v2f b2 = ldb_sw(Dt, kp, n0 + lr + 32);
        v2f b3 = ldb_sw(Dt, kp, n0 + lr + 48);
        acc0 = wmma4(a, b0, acc0);
        acc1 = wmma4(a, b1, acc1);
        acc2 = wmma4(a, b2, acc2);
        acc3 = wmma4(a, b3, acc3);
    }
#pragma unroll
    for (int g = 0; g < 8; ++g) {
        const int row = r0 + g + 8 * half;
        float* o = xio + (size_t)row * 256 + n0 + lr;
        o[ 0] = acc0[g]; o[16] = acc1[g]; o[32] = acc2[g]; o[48] = acc3[g];
    }
}

// Per-chunk local scan: x_t = x_{t-1}@At + v_t  (v_t resident in xio, in-place)
__global__ void k_chunk_scan(float* xio, const float* __restrict__ At) {
    __shared__ float xs[2][16][260];   // stride 260 -> conflict-free frag loads
    const int tid   = threadIdx.x;
    const int wave  = tid >> 5;
    const int lane  = tid & 31;
    const int half  = lane >> 4;
    const int lr    = lane & 15;
    const int n0    = wave * 32;
    const int chunk = blockIdx.x;

    for (int idx = tid; idx < 16 * 260; idx += 256)
        xs[0][idx / 260][idx % 260] = 0.0f;
    __syncthreads();

    int p = 0;
    for (int j = 0; j < CHL; ++j) {
        const int t = chunk * CHL + j;
        v8f acc0 = {}, acc1 = {};
        const float* xrow = &xs[p][lr][0];
#pragma unroll 4
        for (int k0 = 0; k0 < 256; k0 += 4) {
            v2f a = *(const v2f*)(xrow + k0 + 2 * half);
            const int kp = (k0 >> 1) + half;
            v2f b0 = ldb_sw(At, kp, n0 + lr);
            v2f b1 = ldb_sw(At, kp, n0 + lr + 16);
            acc0 = wmma4(a, b0, acc0);
            acc1 = wmma4(a, b1, acc1);
        }
#pragma unroll
        for (int g = 0; g < 8; ++g) {
            const int b = g + 8 * half;
            float* od = xio + ((size_t)b * SEQ + t) * 256 + n0 + lr;
            const float v0 = od[0]  + acc0[g];
            const float v1 = od[16] + acc1[g];
            od[0] = v0; od[16] = v1;
            xs[1 - p][b][n0 + lr]      = v0;
            xs[1 - p][b][n0 + lr + 16] = v1;
        }
        __syncthreads();
        p ^= 1;
    }
}

// Sequential carry over chunk-end states: cs[c] = cs[c-1]@At^CHL + locend[c]
__global__ void k_carry_scan(const float* __restrict__ xio,
                             const float* __restrict__ PwL, float* cs) {
    __shared__ float xs[2][16][260];
    const int tid  = threadIdx.x;
    const int wave = tid >> 5;
    const int lane = tid & 31;
    const int half = lane >> 4;
    const int lr   = lane & 15;
    const int n0   = wave * 32;

    for (int idx = tid; idx < 16 * 260; idx += 256)
        xs[0][idx / 260][idx % 260] = 0.0f;
    __syncthreads();

    int p = 0;
    for (int c = 0; c < NCH; ++c) {
        const int t = c * CHL + (CHL - 1);
        v8f acc0 = {}, acc1 = {};
        const float* xrow = &xs[p][lr][0];
#pragma unroll 4
        for (int k0 = 0; k0 < 256; k0 += 4) {
            v2f a = *(const v2f*)(xrow + k0 + 2 * half);
            const int kp = (k0 >> 1) + half;
            v2f b0 = ldb_sw(PwL, kp, n0 + lr);
            v2f b1 = ldb_sw(PwL, kp, n0 + lr + 16);
            acc0 = wmma4(a, b0, acc0);
            acc1 = wmma4(a, b1, acc1);
        }
#pragma unroll
        for (int g = 0; g < 8; ++g) {
            const int b = g + 8 * half;
            const float* src = xio + ((size_t)b * SEQ + t) * 256 + n0 + lr;
            const float v0 = src[0]  + acc0[g];
            const float v1 = src[16] + acc1[g];
            float* cd = cs + ((size_t)c * 16 + b) * 256 + n0 + lr;
            cd[0] = v0; cd[16] = v1;
            xs[1 - p][b][n0 + lr]      = v0;
            xs[1 - p][b][n0 + lr + 16] = v1;
        }
        __syncthreads();
        p ^= 1;
    }
}

// x_{c*CHL+j} += cs[c-1] @ At^{j+1}   for c >= 1 (fully parallel)
__global__ void k_apply_carry(float* xio, const float* __restrict__ Pw,
                              const float* __restrict__ cs) {
    const int bid  = blockIdx.x;
    const int c    = 1 + bid / CHL;
    const int j    = bid % CHL;
    const int t    = c * CHL + j;
    const int tid  = threadIdx.x;
    const int wave = tid >> 5;
    const int lane = tid & 31;
    const int half = lane >> 4;
    const int lr   = lane & 15;
    const int n0   = wave * 32;

    const float* carry = cs + (size_t)(c - 1) * 16 * 256;
    const float* W     = Pw + (size_t)j * MATN;    // Pw base = At^1 (swizzled)

    v8f acc0 = {}, acc1 = {};
    const float* crow = carry + (size_t)lr * 256;
#pragma unroll 4
    for (int k0 = 0; k0 < 256; k0 += 4) {
        v2f a = *(const v2f*)(crow + k0 + 2 * half);
        const int kp = (k0 >> 1) + half;
        v2f b0 = ldb_sw(W, kp, n0 + lr);
        v2f b1 = ldb_sw(W, kp, n0 + lr + 16);
        acc0 = wmma4(a, b0, acc0);
        acc1 = wmma4(a, b1, acc1);
    }
#pragma unroll
    for (int g = 0; g < 8; ++g) {
        const int b = g + 8 * half;
        float* od = xio + ((size_t)b * SEQ + t) * 256 + n0 + lr;
        od[0]  += acc0[g];
        od[16] += acc1[g];
    }
}

// ---------------- elementwise helpers (stage 0) ----------------
__global__ void k_transpose(const float* in, float* out, float scale) {
    const int idx = blockIdx.x * 256 + threadIdx.x;
    const int i = idx >> 8, j = idx & 255;
    out[(j << 8) + i] = in[idx] * scale;
}
__global__ void k_identity(float* M) {
    const int idx = blockIdx.x * 256 + threadIdx.x;
    M[idx] = ((idx >> 8) == (idx & 255)) ? 1.0f : 0.0f;
}
__global__ void k_axpy(float* O, const float* U, const float* V,
                       float a, float b) {
    const int idx = blockIdx.x * 256 + threadIdx.x;
    O[idx] = a * U[idx] + b * V[idx];
}

// In-place pair-interleave swizzle: row-major -> fragment-native layout.
// Block b owns rows {2b, 2b+1}, whose destination region [b*512, b*512+512)
// is exactly the same memory -> read, sync, write is race-free.
__global__ void k_swz(float* W) {
    float* M = W + (size_t)blockIdx.y * MATN;
    const int r2  = blockIdx.x;
    const int tid = threadIdx.x;
    const float v0 = M[(size_t)(2 * r2)     * 256 + tid];
    const float v1 = M[(size_t)(2 * r2 + 1) * 256 + tid];
    __syncthreads();
    M[(size_t)r2 * 512 + 2 * tid]     = v0;
    M[(size_t)r2 * 512 + 2 * tid + 1] = v1;
}

// ---------------------------------------------------------------------------
extern "C" void kernel_launch(void* const* d_in, const int* in_sizes, int n_in,
                              void* d_out, int out_size, void* d_ws, size_t ws_size,
                              hipStream_t stream) {
    (void)in_sizes; (void)n_in; (void)out_size; (void)ws_size;
    const float* u  = (const float*)d_in[0];
    const float* A  = (const float*)d_in[1];
    const float* Bc = (const float*)d_in[2];
    const float* C  = (const float*)d_in[3];
    const float* D  = (const float*)d_in[4];
    float* out = (float*)d_out;
    float* ws  = (float*)d_ws;

    float* X    = ws;                 // A^T / 64
    float* E0   = ws +  1 * (size_t)MATN;
    float* P0   = ws +  2 * (size_t)MATN;
    float* T0   = ws +  3 * (size_t)MATN;
    float* T1   = ws +  4 * (size_t)MATN;
    float* E1   = ws +  5 * (size_t)MATN;
    float* P1   = ws +  6 * (size_t)MATN;
    float* BcT  = ws +  7 * (size_t)MATN;
    float* Bt   = ws +  8 * (size_t)MATN;
    float* Ct   = ws +  9 * (size_t)MATN;
    float* Dt   = ws + 10 * (size_t)MATN;
    float* cs   = ws + 11 * (size_t)MATN;                 // 128*16*256 = 8 mats
    float* Pw   = ws + 19 * (size_t)MATN;                 // At^1..At^32

    // Stage 0: discretization, all in transposed operator space.
    k_transpose<<<256, 256, 0, stream>>>(A,  X,   1.0f / 64.0f); // X = A^T/2^6
    k_transpose<<<256, 256, 0, stream>>>(Bc, BcT, 1.0f);
    k_transpose<<<256, 256, 0, stream>>>(C,  Ct,  1.0f);
    k_transpose<<<256, 256, 0, stream>>>(D,  Dt,  1.0f);
    k_identity<<<256, 256, 0, stream>>>(E0);
    k_identity<<<256, 256, 0, stream>>>(P0);
    k_identity<<<256, 256, 0, stream>>>(T0);

    float* E = E0; float* Phi = P0; float* T = T0; float* T2 = T1;
    float* Esp = E1; float* Psp = P1;
    for (int k = 1; k <= 12; ++k) {          // Taylor: E=Σ X^k/k!, Phi=Σ X^k/(k+1)!
        k_gemm256<<<8, 256, 0, stream>>>(T2, T, X, nullptr, 1.0f / (float)k, 0.0f);
        k_axpy<<<256, 256, 0, stream>>>(E,   E,   T2, 1.0f, 1.0f);
        k_axpy<<<256, 256, 0, stream>>>(Phi, Phi, T2, 1.0f, 1.0f / (float)(k + 1));
        float* tmp = T; T = T2; T2 = tmp;
    }
    for (int s = 0; s < 6; ++s) {            // squaring: E<-E^2, Phi<-(E+I)Phi/2
        k_gemm256<<<8, 256, 0, stream>>>(Psp, E, Phi, Phi, 0.5f, 0.5f);
        k_gemm256<<<8, 256, 0, stream>>>(Esp, E, E, nullptr, 1.0f, 0.0f);
        float* tE = E; E = Esp; Esp = tE;
        float* tP = Phi; Phi = Psp; Psp = tP;
    }
    // Bt = B_d^T = Bc^T @ phi1(A^T);  At = E = exp(A^T)
    k_gemm256<<<8, 256, 0, stream>>>(Bt, BcT, Phi, nullptr, 1.0f, 0.0f);

    // Powers At^1..At^32 by log-doubling (row-major).
    k_axpy<<<256, 256, 0, stream>>>(Pw, E, E, 1.0f, 0.0f);  // Pw[1] = At
    for (int k = 1; k <= 16; k <<= 1)
        k_powers<<<dim3(8, k), 256, 0, stream>>>(Pw, k);

    // Swizzle all operator matrices into fragment-native layout (in place).
    k_swz<<<dim3(128, 1),  256, 0, stream>>>(E);
    k_swz<<<dim3(128, 1),  256, 0, stream>>>(Bt);
    k_swz<<<dim3(128, 1),  256, 0, stream>>>(Ct);
    k_swz<<<dim3(128, 1),  256, 0, stream>>>(Dt);
    k_swz<<<dim3(128, 32), 256, 0, stream>>>(Pw);

    // Stage 1: v = u @ B_d^T  -> out     (M = 65536 rows)
    k_gemm256_sw<<<2048, 256, 0, stream>>>(out, u, Bt);
    // Stage 2a: per-chunk local scan (in-place)
    k_chunk_scan<<<NCH, 256, 0, stream>>>(out, E);
    // Stage 2b: chunk-boundary carry scan + parallel correction
    k_carry_scan<<<1, 256, 0, stream>>>(out, Pw + (size_t)(CHL - 1) * MATN, cs);
    k_apply_carry<<<(NCH - 1) * CHL, 256, 0, stream>>>(out, Pw, cs);
    // Stage 3: y = x@C^T + u@D^T (in-place over out)
    k_output<<<2048, 256, 0, stream>>>(out, u, Ct, Dt);
}